// Convlayer_33775622815814
// MI455X (gfx1250) — compile-verified
//
#include <hip/hip_runtime.h>

// ---------------------------------------------------------------------------
// PointNet encoder for MI455X (gfx1250, wave32).
//  - 5x fused GEMM+BN+ReLU using V_WMMA_F32_16X16X4_F32 (exact fp32 matrix op)
//  - layers 3..5 write into a [M,896] concat buffer (stride-896 sub-views
//    feed layers 4/5, no concat kernel needed)
//  - FPS parallelized across 16 workgroups per batch (256 blocks total) with
//    a per-round global argmax (packed u64 atomicMax, first-index tie-break)
//    and a monotonic-counter inter-block barrier. Feature tensor (117MB) is
//    L2-resident (192MB L2), so each round is an L2-bandwidth sweep spread
//    over 256 WGPs instead of 16.
//  - final gather emits the [B,896,128] transpose
// ---------------------------------------------------------------------------

typedef __attribute__((ext_vector_type(2))) float v2f;
typedef __attribute__((ext_vector_type(8))) float v8f;

#define BSZ   16
#define NPTS  2048
#define MTOT  (BSZ * NPTS)
#define FDIM  896
#define NSMP  128
#define EPSV  1e-5f
#define GRPB  16                 // blocks per batch for FPS
#define PPB   (NPTS / GRPB)      // 128 points per block

// ---------------------------------------------------------------------------
// GEMM: out[m, n] = relu( (sum_k in[m,k] * W[n,k] + bias[n] - mu[n]) * s + be[n] )
// One wave computes four 16x16 M-tiles for one 16-wide N strip, reusing the
// B (weight) fragment in registers across the four WMMAs per K-step.
// A (16x4 f32): lane holds row = lane%16, k = 2*(lane/16)+j  (j=0,1)
// B (4x16 f32): lane holds col = lane%16, k = 2*(lane/16)+j
// C/D (16x16):  vgpr i, lane: row = i + 8*(lane/16), col = lane%16
// ---------------------------------------------------------------------------
template <int CIN>
__global__ __launch_bounds__(128)
void gemm_bn_relu(const float* __restrict__ in, int ldIn,
                  const float* __restrict__ W,
                  const float* __restrict__ bias,
                  const float* __restrict__ gam,
                  const float* __restrict__ bet,
                  const float* __restrict__ mu,
                  const float* __restrict__ var,
                  float* __restrict__ out, int ldOut)
{
    const int lane  = threadIdx.x & 31;
    const int wave  = threadIdx.x >> 5;
    const int r     = lane & 15;
    const int half  = lane >> 4;
    const int kOff  = half * 2;
    const int mBase = (blockIdx.x * 4 + wave) * 64;
    const int nBase = blockIdx.y * 16;

    const float* a0 = in + (size_t)(mBase +  0 + r) * ldIn;
    const float* a1 = in + (size_t)(mBase + 16 + r) * ldIn;
    const float* a2 = in + (size_t)(mBase + 32 + r) * ldIn;
    const float* a3 = in + (size_t)(mBase + 48 + r) * ldIn;
    const float* bR = W + (size_t)(nBase + r) * CIN;

    v8f acc0 = {}, acc1 = {}, acc2 = {}, acc3 = {};
    constexpr int KP = (CIN + 3) & ~3;

#pragma unroll 8
    for (int k = 0; k < KP; k += 4) {
        const int k0 = k + kOff;
        v2f b, x0, x1, x2, x3;
        if constexpr ((CIN % 4) == 0) {
            b  = *(const v2f*)(bR + k0);
            x0 = *(const v2f*)(a0 + k0);
            x1 = *(const v2f*)(a1 + k0);
            x2 = *(const v2f*)(a2 + k0);
            x3 = *(const v2f*)(a3 + k0);
        } else {                           // K padded with zeros (layer 1, CIN=3)
            const bool e0 = (k0 < CIN), e1 = (k0 + 1 < CIN);
            b[0]  = e0 ? bR[k0] : 0.f;  b[1]  = e1 ? bR[k0 + 1] : 0.f;
            x0[0] = e0 ? a0[k0] : 0.f;  x0[1] = e1 ? a0[k0 + 1] : 0.f;
            x1[0] = e0 ? a1[k0] : 0.f;  x1[1] = e1 ? a1[k0 + 1] : 0.f;
            x2[0] = e0 ? a2[k0] : 0.f;  x2[1] = e1 ? a2[k0 + 1] : 0.f;
            x3[0] = e0 ? a3[k0] : 0.f;  x3[1] = e1 ? a3[k0 + 1] : 0.f;
        }
        acc0 = __builtin_amdgcn_wmma_f32_16x16x4_f32(false, x0, false, b, (short)0, acc0, false, false);
        acc1 = __builtin_amdgcn_wmma_f32_16x16x4_f32(false, x1, false, b, (short)0, acc1, false, false);
        acc2 = __builtin_amdgcn_wmma_f32_16x16x4_f32(false, x2, false, b, (short)0, acc2, false, false);
        acc3 = __builtin_amdgcn_wmma_f32_16x16x4_f32(false, x3, false, b, (short)0, acc3, false, false);
    }

    const int   col = nBase + r;
    const float sc  = gam[col] * rsqrtf(var[col] + EPSV);
    const float sh  = (bias[col] - mu[col]) * sc + bet[col];

    auto emit = [&](const v8f& a, int mt) {
#pragma unroll
        for (int i = 0; i < 8; ++i) {
            const int row = mBase + mt * 16 + i + 8 * half;
            const float val = a[i] * sc + sh;
            out[(size_t)row * ldOut + col] = fmaxf(val, 0.f);
        }
    };
    emit(acc0, 0); emit(acc1, 1); emit(acc2, 2); emit(acc3, 3);
}

// ---------------------------------------------------------------------------
// Per-call zeroing of FPS sync state (graph replays stay deterministic):
// best[b][s] u64 slots (one per round -> no reset protocol), arrive counters,
// and sidx[b][0] = 0.
// ---------------------------------------------------------------------------
__global__ void fps_init(unsigned long long* __restrict__ best,
                         unsigned int* __restrict__ arrive,
                         int* __restrict__ sidx)
{
    const int t = threadIdx.x;
    for (int i = t; i < BSZ * NSMP; i += 256) best[i] = 0ull;
    if (t < BSZ) { arrive[t] = 0u; sidx[t * NSMP] = 0; }
}

// ---------------------------------------------------------------------------
// FPS: 16 blocks per batch, 128 points per block, 8 waves per block.
// Each wave owns one point's 896-float row at a time (coalesced float4 reads).
// Global per-round argmax: packed = (dist_bits << 32) | (0x7fffffff - idx);
// non-negative float bits are order-isomorphic to unsigned, inverted index
// gives jnp.argmax's first-index tie-break. Inter-block barrier = monotonic
// per-batch arrival counter (target 16*s), thread 0 spins with s_sleep.
// ---------------------------------------------------------------------------
__global__ __launch_bounds__(256)
void fps_kernel(const float* __restrict__ feat, int* __restrict__ sidx,
                unsigned long long* __restrict__ best,
                unsigned int* __restrict__ arrive)
{
    const int tid   = threadIdx.x;
    const int lane  = tid & 31;
    const int wave  = tid >> 5;
    const int b     = blockIdx.x / GRPB;
    const int grp   = blockIdx.x % GRPB;
    const int pBase = grp * PPB;
    const float* fb = feat + (size_t)b * NPTS * FDIM;
    unsigned long long* bbest = best + (size_t)b * NSMP;
    unsigned int* barr = arrive + b;

    __shared__ __align__(16) float q[FDIM];
    __shared__ float dists[PPB];
    __shared__ float wd[4];
    __shared__ int   wi[4];
    __shared__ int   sWin;

    for (int p = tid; p < PPB; p += 256) dists[p] = 1e10f;
    int last = 0;

    for (int s = 1; s < NSMP; ++s) {
        // stage query-point features in LDS
        for (int c = tid; c < FDIM; c += 256) q[c] = fb[(size_t)last * FDIM + c];
        __syncthreads();

        // update running min distances; each wave owns 16 points
#pragma unroll 4
        for (int t = 0; t < PPB / 8; ++t) {
            const int li = wave * (PPB / 8) + t;
            const float* row = fb + (size_t)(pBase + li) * FDIM;
            float acc = 0.f;
            for (int c = lane * 4; c < FDIM; c += 128) {   // 7 float4 iters/lane
                const float4 rv = *(const float4*)(row + c);
                const float4 qv = *(const float4*)(q + c);
                const float d0 = rv.x - qv.x, d1 = rv.y - qv.y;
                const float d2 = rv.z - qv.z, d3 = rv.w - qv.w;
                acc += d0 * d0 + d1 * d1 + d2 * d2 + d3 * d3;
            }
#pragma unroll
            for (int off = 16; off > 0; off >>= 1) acc += __shfl_down(acc, off, 32);
            if (lane == 0) dists[li] = fminf(dists[li], acc);
        }
        __syncthreads();

        // block-local argmax over 128 LDS dists (first 4 waves), first-index ties
        float bd = -1.f; int bi = 0x7fffffff;
        if (tid < PPB) { bd = dists[tid]; bi = pBase + tid; }
#pragma unroll
        for (int off = 16; off > 0; off >>= 1) {
            const float od = __shfl_down(bd, off, 32);
            const int   oi = __shfl_down(bi, off, 32);
            if (od > bd || (od == bd && oi < bi)) { bd = od; bi = oi; }
        }
        if (tid < PPB && lane == 0) { wd[wave] = bd; wi[wave] = bi; }
        __syncthreads();

        if (tid == 0) {
            float fbd = wd[0]; int fbi = wi[0];
            for (int w = 1; w < 4; ++w)
                if (wd[w] > fbd || (wd[w] == fbd && wi[w] < fbi)) { fbd = wd[w]; fbi = wi[w]; }

            const unsigned long long packed =
                ((unsigned long long)__float_as_uint(fbd) << 32) |
                (unsigned long long)(unsigned)(0x7fffffff - fbi);
            __hip_atomic_fetch_max(&bbest[s], packed,
                                   __ATOMIC_RELAXED, __HIP_MEMORY_SCOPE_AGENT);
            __hip_atomic_fetch_add(barr, 1u,
                                   __ATOMIC_RELEASE, __HIP_MEMORY_SCOPE_AGENT);

            const unsigned target = (unsigned)GRPB * (unsigned)s;
            while (__hip_atomic_load(barr, __ATOMIC_ACQUIRE,
                                     __HIP_MEMORY_SCOPE_AGENT) < target)
                __builtin_amdgcn_s_sleep(8);

            const unsigned long long w =
                __hip_atomic_load(&bbest[s], __ATOMIC_RELAXED,
                                  __HIP_MEMORY_SCOPE_AGENT);
            const int win = 0x7fffffff - (int)(unsigned)(w & 0xffffffffull);
            sWin = win;
            if (grp == 0) sidx[b * NSMP + s] = win;
        }
        __syncthreads();
        last = sWin;
    }
}

// ---------------------------------------------------------------------------
// out[b, c, s] = feat[b, idx[b,s], c]   (emits the [B,896,128] transpose)
// ---------------------------------------------------------------------------
__global__ __launch_bounds__(256)
void gather_kernel(const float* __restrict__ feat, const int* __restrict__ sidx,
                   float* __restrict__ out)
{
    const int b = blockIdx.x >> 7;
    const int s = blockIdx.x & 127;
    const int p = sidx[b * NSMP + s];
    const float* row = feat + ((size_t)b * NPTS + p) * FDIM;
    float* ob = out + (size_t)b * FDIM * NSMP;
    for (int c = threadIdx.x; c < FDIM; c += 256)
        ob[(size_t)c * NSMP + s] = row[c];
}

// ---------------------------------------------------------------------------
extern "C" void kernel_launch(void* const* d_in, const int* in_sizes, int n_in,
                              void* d_out, int out_size, void* d_ws, size_t ws_size,
                              hipStream_t stream)
{
    (void)in_sizes; (void)n_in; (void)out_size; (void)ws_size;

    const float* x = (const float*)d_in[0];
    const float *W[5], *bi[5], *g[5], *be[5], *mu[5], *vv[5];
    for (int i = 0; i < 5; ++i) {
        const int base = 1 + i * 6;
        W[i]  = (const float*)d_in[base + 0];
        bi[i] = (const float*)d_in[base + 1];
        g[i]  = (const float*)d_in[base + 2];
        be[i] = (const float*)d_in[base + 3];
        mu[i] = (const float*)d_in[base + 4];
        vv[i] = (const float*)d_in[base + 5];
    }

    // Workspace: h1 (8MB) | h2 (8MB) | feat [M,896] (117MB) | sidx | best | arrive
    float* h1   = (float*)d_ws;
    float* h2   = h1 + (size_t)MTOT * 64;
    float* feat = h2 + (size_t)MTOT * 64;
    int*   sidx = (int*)(feat + (size_t)MTOT * FDIM);
    unsigned long long* best = (unsigned long long*)(sidx + BSZ * NSMP);
    unsigned int* arrive = (unsigned int*)(best + BSZ * NSMP);

    const dim3 blk(128);  // 4 waves; each wave: 4 M-tiles x 16 cols; 256 rows/block
    gemm_bn_relu<3>  <<<dim3(128,  4), blk, 0, stream>>>(x,          3,    W[0], bi[0], g[0], be[0], mu[0], vv[0], h1,         64);
    gemm_bn_relu<64> <<<dim3(128,  4), blk, 0, stream>>>(h1,         64,   W[1], bi[1], g[1], be[1], mu[1], vv[1], h2,         64);
    gemm_bn_relu<64> <<<dim3(128,  8), blk, 0, stream>>>(h2,         64,   W[2], bi[2], g[2], be[2], mu[2], vv[2], feat,       FDIM);
    gemm_bn_relu<128><<<dim3(128, 16), blk, 0, stream>>>(feat,       FDIM, W[3], bi[3], g[3], be[3], mu[3], vv[3], feat + 128, FDIM);
    gemm_bn_relu<256><<<dim3(128, 32), blk, 0, stream>>>(feat + 128, FDIM, W[4], bi[4], g[4], be[4], mu[4], vv[4], feat + 384, FDIM);

    fps_init  <<<dim3(1),          dim3(256), 0, stream>>>(best, arrive, sidx);
    fps_kernel<<<dim3(BSZ * GRPB), dim3(256), 0, stream>>>(feat, sidx, best, arrive);
    gather_kernel<<<dim3(BSZ * NSMP), dim3(256), 0, stream>>>(feat, sidx, (float*)d_out);
}